// MaskedMSE_22900765622570
// MI455X (gfx1250) — compile-verified
//
#include <hip/hip_runtime.h>

// MaskedMSE for MI455X (gfx1250, wave32).
// Bandwidth-bound streaming reduction: 151 MB in, 1 float out.
// Roofline: 151 MB / 23.3 TB/s ~= 6.5 us (working set fits the 192 MB L2,
// so default RT cache hints are used to allow L2-resident graph replays).
//
// Structure:
//   - 64 blocks/batch x 256 threads, each thread handles exactly 4 chunks at
//     compile-time stride 16384 -> fully unrolled, branch-free hot loop with
//     36 independent global_load_b128 per thread (max MLP, clause-friendly).
//   - Wave32 reduction via V_WMMA_F32_16X16X4_F32 with an all-ones B matrix:
//     every D element is a row-sum of A replicated across 16 columns, so
//     sum(all D)/16 == sum(all A) regardless of the exact lane striping.
//   - Block combine in LDS, 2 float atomics per block, tiny finalize kernel.

typedef float v2f __attribute__((ext_vector_type(2)));
typedef float v8f __attribute__((ext_vector_type(8)));

#define MM_B      16
#define MM_C      4
#define MM_HW     (512 * 512)
#define MM_NV     (MM_HW / 4)          // float4 vectors per plane = 65536
#define MM_GX     64                   // blocks per batch
#define MM_BS     256                  // threads per block (8 wave32)
#define MM_STRIDE (MM_GX * MM_BS)      // 16384 threads per batch
#define MM_ITERS  (MM_NV / MM_STRIDE)  // 4 chunks per thread (exact)

static_assert(MM_NV % MM_STRIDE == 0, "uniform trip count required");

__global__ void mm_zero_ws(float* __restrict__ ws) {
    unsigned i = threadIdx.x;
    if (i < 2u * MM_B) ws[i] = 0.0f;
}

__global__ __launch_bounds__(MM_BS) void mm_masked_sse(
    const float* __restrict__ in, const float* __restrict__ tg,
    float* __restrict__ ws) {
    const unsigned b   = blockIdx.y;
    const unsigned tid = blockIdx.x * MM_BS + threadIdx.x;   // 0..16383

    const float4* __restrict__ in_b =
        reinterpret_cast<const float4*>(in) + (size_t)b * MM_C * MM_NV;
    const float4* __restrict__ tg_b =
        reinterpret_cast<const float4*>(tg) + (size_t)b * (MM_C + 1) * MM_NV;
    const float4* __restrict__ mk_b = tg_b + (size_t)MM_C * MM_NV;  // mask plane

    // Two interleaved accumulator pairs -> no single serialized FMA chain.
    float sacc[2] = {0.0f, 0.0f};
    float cacc[2] = {0.0f, 0.0f};

#pragma unroll
    for (int k = 0; k < MM_ITERS; ++k) {
        const unsigned i = tid + (unsigned)k * MM_STRIDE;    // compile-time stride
        float4 m = mk_b[i];
        float s = 0.0f;
#pragma unroll
        for (int c = 0; c < MM_C; ++c) {
            float4 a = in_b[(size_t)c * MM_NV + i];
            float4 t = tg_b[(size_t)c * MM_NV + i];
            float dx = a.x - t.x, dy = a.y - t.y, dz = a.z - t.z, dw = a.w - t.w;
            s = fmaf(m.x, dx * dx, s);
            s = fmaf(m.y, dy * dy, s);
            s = fmaf(m.z, dz * dz, s);
            s = fmaf(m.w, dw * dw, s);
        }
        sacc[k & 1] += s;
        cacc[k & 1] += (m.x + m.y) + (m.z + m.w);
    }

    float sum = sacc[0] + sacc[1];
    float cnt = cacc[0] + cacc[1];

    // ---- Wave32 reduction via V_WMMA_F32_16X16X4_F32 ----
    // A is 16x4 f32 (2 VGPRs/lane): partial in A.x, 0 in A.y; B = all ones.
    // Every D element is a row-sum of A replicated across 16 columns, so
    // sum(all 256 D elements) = 16 * sum(all A) = 16 * wave_total.
    // Holds for ANY bijective lane->(m,k) striping of A.
    v2f a_s; a_s.x = sum; a_s.y = 0.0f;
    v2f a_c; a_c.x = cnt; a_c.y = 0.0f;
    v2f ones; ones.x = 1.0f; ones.y = 1.0f;
    v8f zero = {};
    v8f ds = __builtin_amdgcn_wmma_f32_16x16x4_f32(false, a_s, false, ones,
                                                   (short)0, zero, false, false);
    v8f dc = __builtin_amdgcn_wmma_f32_16x16x4_f32(false, a_c, false, ones,
                                                   (short)0, zero, false, false);
    float ts = ((ds[0] + ds[1]) + (ds[2] + ds[3])) + ((ds[4] + ds[5]) + (ds[6] + ds[7]));
    float tc = ((dc[0] + dc[1]) + (dc[2] + dc[3])) + ((dc[4] + dc[5]) + (dc[6] + dc[7]));
#pragma unroll
    for (int off = 16; off > 0; off >>= 1) {
        ts += __shfl_xor(ts, off, 32);
        tc += __shfl_xor(tc, off, 32);
    }
    ts *= (1.0f / 16.0f);   // undo 16x column replication (exact: power of 2)
    tc *= (1.0f / 16.0f);

    // ---- Block combine (8 waves) then 2 atomics per block ----
    __shared__ float red[16];
    const unsigned wave = threadIdx.x >> 5;
    const unsigned lane = threadIdx.x & 31;
    if (lane == 0) { red[2 * wave] = ts; red[2 * wave + 1] = tc; }
    __syncthreads();
    if (threadIdx.x == 0) {
        float s = 0.0f, c = 0.0f;
#pragma unroll
        for (int w = 0; w < 8; ++w) { s += red[2 * w]; c += red[2 * w + 1]; }
        atomicAdd(&ws[2 * b], s);
        atomicAdd(&ws[2 * b + 1], c);
    }
}

__global__ void mm_finalize(const float* __restrict__ ws, float* __restrict__ out) {
    const unsigned lane = threadIdx.x;
    float per = 0.0f;
    if (lane < MM_B) {
        float s = ws[2 * lane];
        float c = ws[2 * lane + 1];
        float denom = c * (float)MM_C;          // C * #masked pixels
        per = (denom > 0.0f) ? (s / denom) : 0.0f;
    }
#pragma unroll
    for (int off = 16; off > 0; off >>= 1) per += __shfl_xor(per, off, 32);
    if (lane == 0) out[0] = per * (1.0f / (float)MM_B);
}

extern "C" void kernel_launch(void* const* d_in, const int* in_sizes, int n_in,
                              void* d_out, int out_size, void* d_ws, size_t ws_size,
                              hipStream_t stream) {
    (void)in_sizes; (void)n_in; (void)out_size; (void)ws_size;
    const float* in = (const float*)d_in[0];   // [16,4,512,512] f32
    const float* tg = (const float*)d_in[1];   // [16,5,512,512] f32
    float* out = (float*)d_out;                // 1 float
    float* ws  = (float*)d_ws;                 // 32 floats: (sum, cnt) per batch

    mm_zero_ws<<<1, 32, 0, stream>>>(ws);
    dim3 grid(MM_GX, MM_B, 1);
    mm_masked_sse<<<grid, MM_BS, 0, stream>>>(in, tg, ws);
    mm_finalize<<<1, 32, 0, stream>>>(ws, out);
}